// OPTSelfAttention_50156628083009
// MI455X (gfx1250) — compile-verified
//
#include <hip/hip_runtime.h>
#include <math.h>

typedef float v2f __attribute__((ext_vector_type(2)));
typedef float v8f __attribute__((ext_vector_type(8)));

#define HDIM_TOT 4096      // H
#define NQKV     12288     // 3*H
#define NHEAD    32
#define HD       128       // head_dim
#define BSZ      16
#define SPAST    2048
#define STOT     2049

// ---------------------------------------------------------------------------
// Kernel 1: qkv(16 x 12288) = hidden(16 x 4096) @ W(4096 x 12288)  (+bias)
// using V_WMMA_F32_16X16X4_F32.  Each wave owns one 16-column strip and
// marches K in steps of 4.  hidden is staged in padded LDS (bank-conflict
// free: 516 % 64 = 4 -> 32 distinct banks across the wave).
// Epilogue de-interleaves q/k/v (col%3) and adds bias; k_new/v_new go
// straight into d_out (they ARE two of the reference outputs).
// ---------------------------------------------------------------------------
#define KCHUNK  512
#define LDS_PAD 516
#define WAVES1  4

__global__ __launch_bounds__(32 * WAVES1)
void qkv_wmma_kernel(const float* __restrict__ hidden,
                     const float* __restrict__ W,
                     const float* __restrict__ bias,
                     float* __restrict__ q_ws,
                     float* __restrict__ out)
{
    __shared__ float sA[16 * LDS_PAD];

    const int tid  = threadIdx.x;
    const int wave = tid >> 5;
    const int lane = tid & 31;
    const int half = lane >> 4;      // 0: lanes 0-15, 1: lanes 16-31
    const int l16  = lane & 15;
    const int n0   = (blockIdx.x * WAVES1 + wave) * 16;
    const int offK = half * 2;       // ISA layout: upper lane half holds K+2

    v8f c = {};

    for (int k0 = 0; k0 < HDIM_TOT; k0 += KCHUNK) {
        // cooperative load: hidden[0:16, k0:k0+512] -> padded LDS
        #pragma unroll
        for (int i = 0; i < (16 * KCHUNK) / (WAVES1 * 32 * 4); ++i) {
            int f   = (tid + i * WAVES1 * 32) * 4;
            int m   = f / KCHUNK;
            int col = f % KCHUNK;
            float4 v = *(const float4*)(hidden + m * HDIM_TOT + k0 + col);
            *(float4*)(&sA[m * LDS_PAD + col]) = v;
        }
        __syncthreads();

        #pragma unroll 4
        for (int kk = 0; kk < KCHUNK; kk += 4) {
            v2f a, b;
            // A 16x4 f32: VGPR0 = K=0|K=2 (lane halves), VGPR1 = K=1|K=3
            a.x = sA[l16 * LDS_PAD + kk + offK];
            a.y = sA[l16 * LDS_PAD + kk + offK + 1];
            // B 4x16 f32: row K striped across 16 lanes, halves split K
            const float* wp = W + (size_t)(k0 + kk + offK) * NQKV + n0 + l16;
            b.x = wp[0];
            b.y = wp[NQKV];
            // (neg_a, A, neg_b, B, c_mod, C, reuse_a, reuse_b)
            c = __builtin_amdgcn_wmma_f32_16x16x4_f32(
                    false, a, false, b, (short)0, c, false, false);
        }
        __syncthreads();
    }

    // epilogue: C layout  lane -> N = n0 + l16,  VGPR j -> M = j + half*8
    const int col   = n0 + l16;
    const int which = col % 3;           // 0=q, 1=k_new, 2=v_new
    const int hd    = col / 3;           // h*128 + d
    const float bv  = bias[which * 4096 + hd];
    #pragma unroll
    for (int j = 0; j < 8; ++j) {
        int   m   = j + half * 8;        // batch row
        float val = c[j] + bv;
        if (which == 0)      q_ws[m * HDIM_TOT + hd]          = val;
        else if (which == 1) out[65536  + m * HDIM_TOT + hd]  = val; // key_states
        else                 out[131072 + m * HDIM_TOT + hd]  = val; // value_states
    }
}

// ---------------------------------------------------------------------------
// Kernel 2: decode attention.  One workgroup per (b,h); 8 waves.
//   A: scores[s] = scale * q . K[s]   (coalesced 512B K-row per wave + shuffle)
//   B: block softmax in LDS
//   C: attn = sum_s p_s * V[s]        (s split over waves, float4 per lane)
// ---------------------------------------------------------------------------
#define WAVES2 8

__global__ __launch_bounds__(32 * WAVES2)
void attn_kernel(const float* __restrict__ q_ws,
                 const float* __restrict__ cache_key,
                 const float* __restrict__ cache_value,
                 float* __restrict__ out)
{
    __shared__ float s_scores[STOT];
    __shared__ float s_red[32 * WAVES2];
    __shared__ float s_part[WAVES2][HD];

    const int b    = blockIdx.x >> 5;
    const int h    = blockIdx.x & 31;
    const int tid  = threadIdx.x;
    const int wave = tid >> 5;
    const int lane = tid & 31;

    const float scale = 0.08838834764831845f;   // 1/sqrt(128)

    const float* qrow = q_ws + b * HDIM_TOT + h * HD;
    float4 q4 = *(const float4*)(qrow + lane * 4);

    const float* knew = out + 65536  + b * HDIM_TOT + h * HD;
    const float* vnew = out + 131072 + b * HDIM_TOT + h * HD;

    // ---- Phase A: scores -------------------------------------------------
    for (int s = wave; s < STOT; s += WAVES2) {
        const float* krow = (s < SPAST)
            ? cache_key + (((size_t)b * SPAST + s) * NHEAD + h) * HD
            : knew;
        float4 k4 = *(const float4*)(krow + lane * 4);
        float d = q4.x * k4.x + q4.y * k4.y + q4.z * k4.z + q4.w * k4.w;
        d += __shfl_down(d, 16);
        d += __shfl_down(d, 8);
        d += __shfl_down(d, 4);
        d += __shfl_down(d, 2);
        d += __shfl_down(d, 1);
        if (lane == 0) s_scores[s] = d * scale;
    }
    __syncthreads();

    // ---- Phase B: softmax ------------------------------------------------
    float mx = -3.0e38f;
    for (int s = tid; s < STOT; s += 32 * WAVES2) mx = fmaxf(mx, s_scores[s]);
    s_red[tid] = mx;
    __syncthreads();
    for (int off = 16 * WAVES2; off > 0; off >>= 1) {
        if (tid < off) s_red[tid] = fmaxf(s_red[tid], s_red[tid + off]);
        __syncthreads();
    }
    mx = s_red[0];
    __syncthreads();

    float sum = 0.f;
    for (int s = tid; s < STOT; s += 32 * WAVES2) {
        float e = __expf(s_scores[s] - mx);
        s_scores[s] = e;
        sum += e;
    }
    s_red[tid] = sum;
    __syncthreads();
    for (int off = 16 * WAVES2; off > 0; off >>= 1) {
        if (tid < off) s_red[tid] += s_red[tid + off];
        __syncthreads();
    }
    const float inv = 1.0f / s_red[0];
    __syncthreads();

    // ---- Phase C: attn = probs . V --------------------------------------
    float ax = 0.f, ay = 0.f, az = 0.f, aw = 0.f;
    for (int s = wave; s < STOT; s += WAVES2) {
        const float* vrow = (s < SPAST)
            ? cache_value + (((size_t)b * SPAST + s) * NHEAD + h) * HD
            : vnew;
        if (s + WAVES2 < SPAST) {
            // next V row for this wave -> global_prefetch_b8
            __builtin_prefetch(vrow + (size_t)WAVES2 * NHEAD * HD, 0, 1);
        }
        float4 v4 = *(const float4*)(vrow + lane * 4);
        float p = s_scores[s] * inv;
        ax += p * v4.x; ay += p * v4.y; az += p * v4.z; aw += p * v4.w;
    }
    s_part[wave][lane * 4 + 0] = ax;
    s_part[wave][lane * 4 + 1] = ay;
    s_part[wave][lane * 4 + 2] = az;
    s_part[wave][lane * 4 + 3] = aw;
    __syncthreads();

    if (tid < HD) {
        float r = 0.f;
        #pragma unroll
        for (int w = 0; w < WAVES2; ++w) r += s_part[w][tid];
        out[b * HDIM_TOT + h * HD + tid] = r;   // attn_output
    }
}

// ---------------------------------------------------------------------------
extern "C" void kernel_launch(void* const* d_in, const int* in_sizes, int n_in,
                              void* d_out, int out_size, void* d_ws, size_t ws_size,
                              hipStream_t stream) {
    const float* hidden  = (const float*)d_in[0];   // (16, 4096)
    const float* cache_k = (const float*)d_in[1];   // (16, 2048, 32, 128)
    const float* cache_v = (const float*)d_in[2];   // (16, 2048, 32, 128)
    const float* W       = (const float*)d_in[3];   // (4096, 12288)
    const float* bias    = (const float*)d_in[4];   // (3, 32, 128)

    float* out  = (float*)d_out;   // [attn 65536 | key 65536 | value 65536]
    float* q_ws = (float*)d_ws;    // 16*4096 floats = 256 KB scratch

    qkv_wmma_kernel<<<NQKV / (16 * WAVES1), 32 * WAVES1, 0, stream>>>(
        hidden, W, bias, q_ws, out);
    attn_kernel<<<BSZ * NHEAD, 32 * WAVES2, 0, stream>>>(
        q_ws, cache_k, cache_v, out);
}